// MySTFT_62053687492959
// MI455X (gfx1250) — compile-verified
//
#include <hip/hip_runtime.h>

typedef float v2f __attribute__((ext_vector_type(2)));
typedef float v8f __attribute__((ext_vector_type(8)));

#define N_TIME   65536
#define N_OUT_T  65531     // N_TIME - N_FFT + 1
#define N_BATCH  256
#define TILES_PER_ROW 4096 // 65536 / 16, last tile partially valid

__global__ __launch_bounds__(256) void stft_wmma_f32_kernel(
    const float* __restrict__ x,
    const float* __restrict__ win,
    float* __restrict__ out)
{
    const int lane = threadIdx.x & 31;
    const int wave = threadIdx.x >> 5;
    const int half = lane >> 4;   // lane half selects high K bits / high M bits
    const int n16  = lane & 15;

    // ---- Build A fragments (coefficient matrix, 16 rows x 4 taps) ----
    // Row c (c<8): c = 2*f + isIm ; f = frequency bin. Rows 8..15 are zero.
    // A-matrix VGPR layout (32-bit, 16x4): lane L<16 holds A[L][0],A[L][1];
    // lane L>=16 holds A[L-16][2],A[L-16][3]  =>  K = vgpr + 2*half.
    float w0 = win[0], w1 = win[1], w2 = win[2], w3 = win[3], w4 = win[4], w5 = win[5];
    float wv[8] = {w0, w1, w2, w3, w4, w5, 0.f, 0.f};

    const int c = n16;          // channel row of A for this lane
    const int f = c >> 1;
    const bool isIm = (c & 1) != 0;
    const float ANG = 1.0471975511965977f; // 2*pi / 6

    auto coef = [&](int tap) -> float {
        if (c >= 8 || tap >= 6) return 0.0f;
        float ang = ANG * (float)f * (float)tap;
        float v = isIm ? (-wv[tap] * __sinf(ang)) : (wv[tap] * __cosf(ang));
        return v;
    };

    v2f a0, a1;
    a0.x = coef(2 * half);      // taps 0..3
    a0.y = coef(2 * half + 1);
    a1.x = coef(4 + 2 * half);  // taps 4..7 (6,7 -> 0)
    a1.y = coef(5 + 2 * half);

    // ---- Grid-stride loop over (batch, time-tile) ----
    const long nTiles     = (long)N_BATCH * TILES_PER_ROW;
    const long totalWaves = (long)gridDim.x * (blockDim.x >> 5);
    long tile = (long)blockIdx.x * (blockDim.x >> 5) + wave;

    for (; tile < nTiles; tile += totalWaves) {
        const int b  = (int)(tile >> 12);       // / 4096
        const int ti = (int)(tile & 4095);
        const int t0 = ti << 4;                 // 16 time positions per tile
        const float* __restrict__ xrow = x + (long)b * N_TIME;

        // B fragments: B[k][n] = x[t0 + n + k], K = vgpr + 2*half -> each lane
        // loads a contiguous pair starting at t0 + n + 2*half (and +4 for taps 4..7).
        int i0 = t0 + n16 + 2 * half;
        int i1 = i0 + 4;
        if (i0 > N_TIME - 2) i0 = N_TIME - 2;   // clamp OOB (coeff is 0 there anyway
        if (i1 > N_TIME - 2) i1 = N_TIME - 2;   //  or the store is masked)

        v2f b0, b1;
        b0.x = xrow[i0];
        b0.y = xrow[i0 + 1];
        b1.x = xrow[i1];
        b1.y = xrow[i1 + 1];

        // ---- D = A0*B0 + A1*B1 : two chained fp32 WMMAs (uniform EXEC) ----
        v8f acc = {};
        acc = __builtin_amdgcn_wmma_f32_16x16x4_f32(
            false, a0, false, b0, (short)0, acc, false, false);
        acc = __builtin_amdgcn_wmma_f32_16x16x4_f32(
            false, a1, false, b1, (short)0, acc, false, false);

        // ---- Store: VGPR r = channel r, lane (0..15) = time t0+lane.
        // Channel pair (2f, 2f+1) = (re_f, im_f) -> contiguous (re,im) in output.
        const int t = t0 + n16;
        if (half == 0 && t < N_OUT_T) {
            #pragma unroll
            for (int fo = 0; fo < 4; ++fo) {
                v2f pair;
                pair.x = acc[2 * fo];      // re
                pair.y = acc[2 * fo + 1];  // im
                float* p = out + ((((long)b * 4 + fo) * N_OUT_T + t) << 1);
                __builtin_nontemporal_store(pair, (v2f*)p); // streamed 537MB output
            }
        }
    }
}

extern "C" void kernel_launch(void* const* d_in, const int* in_sizes, int n_in,
                              void* d_out, int out_size, void* d_ws, size_t ws_size,
                              hipStream_t stream)
{
    const float* x   = (const float*)d_in[0];
    const float* win = (const float*)d_in[1];
    float* out       = (float*)d_out;

    dim3 block(256);   // 8 waves per block (wave32)
    dim3 grid(8192);   // 65536 waves -> 16 tiles per wave grid-stride
    stft_wmma_f32_kernel<<<grid, block, 0, stream>>>(x, win, out);
}